// MultiTimeAttention_1984274890993
// MI455X (gfx1250) — compile-verified
//
#include <hip/hip_runtime.h>
#include <math.h>

typedef __attribute__((ext_vector_type(2))) float v2f;
typedef __attribute__((ext_vector_type(8))) float v8f;
typedef __attribute__((ext_vector_type(4))) int   v4i;

#define L_SEQ 2048
#define NHEAD 8
#define DK    16
#define DV    128
#define BROWS 8192   // BS * L_SEQ

#if __has_builtin(__builtin_amdgcn_global_load_async_to_lds_b128) && \
    __has_builtin(__builtin_amdgcn_global_load_async_to_lds_b32) &&  \
    __has_builtin(__builtin_amdgcn_s_wait_asynccnt)
#define HAVE_ASYNC 1
#define NBUF 2
#else
#define HAVE_ASYNC 0
#define NBUF 1
#endif

__device__ __forceinline__ v8f wmma4(v2f a, v2f b, v8f c) {
  // D = A(16x4 f32) * B(4x16 f32) + C(16x16 f32)
  return __builtin_amdgcn_wmma_f32_16x16x4_f32(false, a, false, b, (short)0, c,
                                               false, false);
}

#if HAVE_ASYNC
__device__ __forceinline__ void async_cp16(const float* g, float* l) {
  __builtin_amdgcn_global_load_async_to_lds_b128(
      (v4i*)(float*)g,
      (__attribute__((address_space(3))) v4i*)l, 0, 0);
}
__device__ __forceinline__ void async_cp4(const float* g, float* l) {
  __builtin_amdgcn_global_load_async_to_lds_b32(
      (int*)(float*)g,
      (__attribute__((address_space(3))) int*)l, 0, 0);
}
#endif

// C = A(M,K) @ W(N,K)^T + bias.  One 16x16 tile per wave, 4 waves per block.
// MODE 0: C row-major (M,N).  MODE 1: head-split store into (b, h, L, 16).
template <int MODE>
__global__ __launch_bounds__(128) void gemm_xwT_bias(
    const float* __restrict__ A, const float* __restrict__ W,
    const float* __restrict__ bias, float* __restrict__ C,
    int M, int N, int K) {
  const int lane = threadIdx.x & 31;
  const int widx = threadIdx.x >> 5;
  const int lm   = lane & 15;
  const int hf   = lane >> 4;        // 0: lanes 0-15, 1: lanes 16-31
  const int ntn  = N >> 4;
  const int tile = blockIdx.x * 4 + widx;
  const int tm   = tile / ntn;
  const int tn   = tile % ntn;

  const float* arow = A + (size_t)(tm * 16 + lm) * K + 2 * hf;
  const float* wrow = W + (size_t)(tn * 16 + lm) * K + 2 * hf;

  v8f acc = {};
#pragma unroll 4
  for (int kk = 0; kk < K; kk += 4) {
    v2f a = *(const v2f*)(arow + kk);   // A: 16x4, lanes 16-31 hold K+2..3
    v2f b = *(const v2f*)(wrow + kk);   // B = W^T: 4x16, same half split
    acc = wmma4(a, b, acc);
  }

  const float bb = bias[tn * 16 + lm];
#pragma unroll
  for (int e = 0; e < 8; ++e) {
    const int row = tm * 16 + e + hf * 8;   // C layout: VGPR e -> M=e (+8 upper)
    const int col = tn * 16 + lm;
    const float val = acc[e] + bb;
    if (MODE == 0) {
      C[(size_t)row * N + col] = val;
    } else {
      const int bidx = row >> 11, l = row & 2047;
      const int hh = col >> 4, dd = col & 15;
      C[(((size_t)bidx * NHEAD + hh) * L_SEQ + l) * DK + dd] = val;
    }
  }
}

__global__ __launch_bounds__(256) void maskbias_kernel(
    const int* __restrict__ mask, float* __restrict__ mb, int n) {
  int i = blockIdx.x * 256 + threadIdx.x;
  if (i < n) mb[i] = mask[i] ? 0.0f : -1e9f;
}

// Flash attention, fp32 WMMA. Grid: (L/64, B*H). Block: 128 (4 waves).
// Each wave owns 16 queries; WG stages 64 keys of K-head and V in LDS.
// With HAVE_ASYNC: double-buffered async global->LDS copies (ASYNCcnt),
// stage s+1 streams in while WMMAs process stage s.
__global__ __launch_bounds__(128) void attn_kernel(
    const float* __restrict__ qh, const float* __restrict__ kh,
    const float* __restrict__ v, const float* __restrict__ mb,
    float* __restrict__ ctx) {
  __shared__ float lds_v[NBUF][64 * 136];  // V tile row-major [key][d], pad 136
  __shared__ float lds_k[NBUF][64 * 18];   // K tile [key][dk], pad 18
  __shared__ float lds_mb[NBUF][64];

  const int tid  = threadIdx.x;
  const int lane = tid & 31;
  const int widx = tid >> 5;
  const int lm   = lane & 15;
  const int hf   = lane >> 4;
  const int bh   = blockIdx.y;
  const int b    = bh >> 3;
  const int h    = bh & 7;
  const int q0   = blockIdx.x * 64 + widx * 16;

  const float* vbase = v + (size_t)b * L_SEQ * DV;
  const float* kbase = kh + (size_t)bh * L_SEQ * DK;
  const float* mbase = mb + b * L_SEQ;

  // Q tile (16 q x 16 dk) preloaded as WMMA-B operands for S^T = K * Q^T.
  v2f qb[4];
#pragma unroll
  for (int s = 0; s < 4; ++s)
    qb[s] = *(const v2f*)(qh + ((size_t)bh * L_SEQ + q0 + lm) * DK + s * 4 + 2 * hf);

  v8f o[8] = {};                 // O^T accumulators: 8 d-tiles of 16, q = lane
  float m_run = -3.0e38f;
  float l_run = 0.0f;

  const int nstages = L_SEQ / 64;

#if HAVE_ASYNC
  // Prologue: stage 0 into buffer 0.  19 async ops per wave per stage.
  {
#pragma unroll 4
    for (int i = tid; i < 64 * 32; i += 128) {
      const int key = i >> 5, dq = (i & 31) << 2;
      async_cp16(vbase + (size_t)key * DV + dq, &lds_v[0][key * 136 + dq]);
    }
#pragma unroll
    for (int i = tid; i < 64 * 4; i += 128) {
      const int key = i >> 2, dq = (i & 3) << 2;
      async_cp16(kbase + (size_t)key * DK + dq, &lds_k[0][key * 18 + dq]);
    }
    const int i = tid & 63;
    async_cp4(mbase + i, &lds_mb[0][i]);
  }
#endif

  for (int si = 0; si < nstages; ++si) {
    const int s0  = si * 64;
    const int cur = HAVE_ASYNC ? (si & 1) : 0;
    __syncthreads();   // all waves finished reading the buffer we now refill

#if HAVE_ASYNC
    if (si + 1 < nstages) {
      const int nxt = cur ^ 1;
      const int n0  = s0 + 64;
#pragma unroll 4
      for (int i = tid; i < 64 * 32; i += 128) {
        const int key = i >> 5, dq = (i & 31) << 2;
        async_cp16(vbase + (size_t)(n0 + key) * DV + dq,
                   &lds_v[nxt][key * 136 + dq]);
      }
#pragma unroll
      for (int i = tid; i < 64 * 4; i += 128) {
        const int key = i >> 2, dq = (i & 3) << 2;
        async_cp16(kbase + (size_t)(n0 + key) * DK + dq,
                   &lds_k[nxt][key * 18 + dq]);
      }
      const int i = tid & 63;
      async_cp4(mbase + n0 + i, &lds_mb[nxt][i]);
      __builtin_amdgcn_s_wait_asynccnt(19);  // stage si landed; si+1 in flight
    } else {
      __builtin_amdgcn_s_wait_asynccnt(0);
    }
#else
    // Manual staging fallback.
    for (int i = tid; i < 64 * 32; i += 128) {
      const int key = i >> 5, dq = (i & 31) << 2;
      const float4 val = *(const float4*)(vbase + (size_t)(s0 + key) * DV + dq);
      *(float4*)&lds_v[0][key * 136 + dq] = val;
    }
    for (int i = tid; i < 64 * 4; i += 128) {
      const int key = i >> 2, dq = (i & 3) << 2;
      const float4 val = *(const float4*)(kbase + (size_t)(s0 + key) * DK + dq);
      lds_k[0][key * 18 + dq + 0] = val.x;
      lds_k[0][key * 18 + dq + 1] = val.y;
      lds_k[0][key * 18 + dq + 2] = val.z;
      lds_k[0][key * 18 + dq + 3] = val.w;
    }
    if (tid < 64) lds_mb[0][tid] = mbase[s0 + tid];
#endif
    __syncthreads();   // publish stage si to all waves

#pragma unroll
    for (int kt = 0; kt < 4; ++kt) {
      const int kl = kt * 16;
      // S^T (16 keys x 16 queries): A = K tile, B = Q^T, K-dim = d_k = 16.
      v8f sacc = {};
#pragma unroll
      for (int s = 0; s < 4; ++s) {
        v2f ka = *(const v2f*)&lds_k[cur][(kl + lm) * 18 + s * 4 + 2 * hf];
        sacc = wmma4(ka, qb[s], sacc);
      }
      // Column (per-query) online softmax; key = slot e (+8 upper half).
      float p[8];
      float mt = -3.0e38f;
#pragma unroll
      for (int e = 0; e < 8; ++e) {
        const float sv = sacc[e] * 0.25f + lds_mb[cur][kl + e + hf * 8];
        p[e] = sv;
        mt = fmaxf(mt, sv);
      }
      mt = fmaxf(mt, __shfl_xor(mt, 16));
      const float mnew = fmaxf(m_run, mt);
      const float fsc  = __expf(m_run - mnew);
      m_run = mnew;
      float ls = 0.0f;
#pragma unroll
      for (int e = 0; e < 8; ++e) { p[e] = __expf(p[e] - mnew); ls += p[e]; }
      ls += __shfl_xor(ls, 16);
      l_run = l_run * fsc + ls;
#pragma unroll
      for (int t = 0; t < 8; ++t) o[t] *= fsc;

      // O^T += V^T * P^T.  Build B operand from P^T registers (2 shuffles/step).
#pragma unroll
      for (int s = 0; s < 4; ++s) {
        const int base = (s & 1) * 4;
        const float xa = p[base + 0], xb = p[base + 1];
        const float xc = p[base + 2], xd = p[base + 3];
        float b0, b1;
        if (s < 2) {            // keys base..base+3 live in lower half
          const float yc = __shfl_xor(xc, 16), yd = __shfl_xor(xd, 16);
          b0 = hf ? yc : xa;
          b1 = hf ? yd : xb;
        } else {                // keys 8+base.. live in upper half
          const float ya = __shfl_xor(xa, 16), yb = __shfl_xor(xb, 16);
          b0 = hf ? xc : ya;
          b1 = hf ? xd : yb;
        }
        v2f pb; pb.x = b0; pb.y = b1;
        const int kkey = kl + s * 4 + 2 * hf;
#pragma unroll
        for (int t = 0; t < 8; ++t) {
          const int d = t * 16 + lm;
          v2f av;
          av.x = lds_v[cur][(kkey + 0) * 136 + d];
          av.y = lds_v[cur][(kkey + 1) * 136 + d];
          o[t] = wmma4(av, pb, o[t]);
        }
      }
    }
  }

  const float inv = 1.0f / l_run;
  float* crow = ctx + ((size_t)b * L_SEQ + q0 + lm) * (NHEAD * DV) + h * DV;
#pragma unroll
  for (int t = 0; t < 8; ++t)
#pragma unroll
    for (int e = 0; e < 8; ++e)
      crow[t * 16 + e + hf * 8] = o[t][e] * inv;
}

extern "C" void kernel_launch(void* const* d_in, const int* in_sizes, int n_in,
                              void* d_out, int out_size, void* d_ws, size_t ws_size,
                              hipStream_t stream) {
  const float* q    = (const float*)d_in[0];
  const float* k    = (const float*)d_in[1];
  const float* v    = (const float*)d_in[2];
  const int*   mask = (const int*)d_in[3];
  const float* Wq   = (const float*)d_in[4];
  const float* bq   = (const float*)d_in[5];
  const float* Wk   = (const float*)d_in[6];
  const float* bk   = (const float*)d_in[7];
  const float* Wo   = (const float*)d_in[8];
  const float* bo   = (const float*)d_in[9];
  float* out = (float*)d_out;

  float* ws  = (float*)d_ws;
  float* qh  = ws;                         // (B,H,L,16)   1,048,576 f
  float* kh  = ws + 1048576;               // (B,H,L,16)   1,048,576 f
  float* mbf = ws + 2097152;               // (B,L)        8,192 f
  float* ctx = ws + 2105344;               // (B*L, 1024)  8,388,608 f
  (void)in_sizes; (void)n_in; (void)out_size; (void)ws_size;

  const dim3 blk(128);
  // Q/K projections: (8192,128)@(128,128)^T -> head-split layout.
  const int proj_blocks = (BROWS / 16) * (128 / 16) / 4;   // 1024
  gemm_xwT_bias<1><<<proj_blocks, blk, 0, stream>>>(q, Wq, bq, qh, BROWS, 128, 128);
  gemm_xwT_bias<1><<<proj_blocks, blk, 0, stream>>>(k, Wk, bk, kh, BROWS, 128, 128);

  maskbias_kernel<<<(BROWS + 255) / 256, 256, 0, stream>>>(mask, mbf, BROWS);

  attn_kernel<<<dim3(L_SEQ / 64, 4 * NHEAD), blk, 0, stream>>>(qh, kh, v, mbf, ctx);

  // Output projection: (8192,1024)@(512,1024)^T + bo.
  const int out_blocks = (BROWS / 16) * (512 / 16) / 4;    // 4096
  gemm_xwT_bias<0><<<out_blocks, blk, 0, stream>>>(ctx, Wo, bo, out, BROWS, 512, 1024);
}